// GraphAttentionLayer_15917148799276
// MI455X (gfx1250) — compile-verified
//
#include <hip/hip_runtime.h>
#include <cstdint>

typedef float v2f __attribute__((ext_vector_type(2)));
typedef float v8f __attribute__((ext_vector_type(8)));

#define LDSTRIDE 132   // 128 + 4 pad: (n*132 + k) mod 64 = (4n + k) mod 64 -> conflict-free fragments
#define NF 128

// One workgroup per (b,s) slice: full 128x128 GAT step in LDS.
// Layouts: sA row-major (h, then attention). sWT / sWhT are K-contiguous
// (transposed) so every WMMA B fragment is one aligned ds_load_b64.
__global__ __launch_bounds__(256) void gat_fused_kernel(
    const float* __restrict__ h,    // (BS, 128, 128)
    const float* __restrict__ Ww,   // (128, 128)
    const float* __restrict__ Wb,   // (128)
    const float* __restrict__ a1,   // (128)
    const float* __restrict__ a2,   // (128)
    const float* __restrict__ ab,   // scalar
    const int*   __restrict__ adj,  // (128, 128)
    float* __restrict__ out)        // (BS, 128, 128)
{
    __shared__ float sA  [NF * LDSTRIDE];  // h tile, later attention (row-major)
    __shared__ float sWT [NF * LDSTRIDE];  // W_w transposed:  sWT[n*S + k]  = W[k][n]
    __shared__ float sWhT[NF * LDSTRIDE];  // Wh  transposed:  sWhT[n*S + m] = Wh[m][n]
    __shared__ float sf1[NF];
    __shared__ float sf2[NF];

    const int tid  = threadIdx.x;
    const int wave = tid >> 5;
    const int lane = tid & 31;
    const int l16  = lane & 15;
    const int half = lane >> 4;          // 0 or 1
    const int kb   = half * 2;           // K sub-offset for 16x16x4 fragments

    const size_t bs = blockIdx.x;
    const float* hp = h   + bs * (size_t)(NF * NF);
    float*       op = out + bs * (size_t)(NF * NF);

    // ---- stage h (row-major) and W_w (transposed) into LDS ----
    for (int idx = tid; idx < (NF * NF) / 4; idx += 256) {
        int row = idx >> 5;              // 32 float4 per row
        int c4  = (idx & 31) << 2;
        float4 hv = *(const float4*)(hp + row * NF + c4);
        *(float4*)(&sA[row * LDSTRIDE + c4]) = hv;
        float4 wv = *(const float4*)(Ww + row * NF + c4);
        sWT[(c4 + 0) * LDSTRIDE + row] = wv.x;
        sWT[(c4 + 1) * LDSTRIDE + row] = wv.y;
        sWT[(c4 + 2) * LDSTRIDE + row] = wv.z;
        sWT[(c4 + 3) * LDSTRIDE + row] = wv.w;
    }
    __syncthreads();

    // ---- GEMM1: Wh = h @ W_w + W_b  (V_WMMA_F32_16X16X4_F32) ----
    {
        const int mt = wave;                 // tile row (16 rows per wave)
        const int m  = mt * 16 + l16;        // A-fragment row for this lane
        v8f acc[8];
        #pragma unroll
        for (int nt = 0; nt < 8; ++nt) {
            float bias = Wb[nt * 16 + l16];
            #pragma unroll
            for (int r = 0; r < 8; ++r) acc[nt][r] = bias;
        }
        for (int k0 = 0; k0 < NF; k0 += 4) {
            v2f a = *(const v2f*)(&sA[m * LDSTRIDE + k0 + kb]);   // one ds_load_b64
            #pragma unroll
            for (int nt = 0; nt < 8; ++nt) {
                int n = nt * 16 + l16;
                v2f b = *(const v2f*)(&sWT[n * LDSTRIDE + k0 + kb]);  // one ds_load_b64
                acc[nt] = __builtin_amdgcn_wmma_f32_16x16x4_f32(
                    false, a, false, b, (short)0, acc[nt], false, false);
            }
        }
        // D layout: VGPR r -> row mt*16 + r + 8*half, col n. In transposed LDS the
        // 8 components per lane are CONTIGUOUS -> two b128 stores per tile.
        #pragma unroll
        for (int nt = 0; nt < 8; ++nt) {
            int n = nt * 16 + l16;
            float* dst = &sWhT[n * LDSTRIDE + mt * 16 + half * 8];
            *(float4*)(dst)     = make_float4(acc[nt][0], acc[nt][1], acc[nt][2], acc[nt][3]);
            *(float4*)(dst + 4) = make_float4(acc[nt][4], acc[nt][5], acc[nt][6], acc[nt][7]);
        }
    }
    __syncthreads();

    // ---- f1 = Wh@a1 + a_b, f2 = Wh@a2 (lane-consecutive, conflict-free) ----
    {
        int r = tid & 127;
        const float* av = (tid < 128) ? a1 : a2;
        float s = 0.f;
        for (int n = 0; n < NF; ++n) s += sWhT[n * LDSTRIDE + r] * av[n];
        if (tid < 128) sf1[r] = s + ab[0];
        else           sf2[r] = s;
    }
    __syncthreads();

    // ---- e = leaky_relu(f1[i]+f2[j]+a_b), mask, softmax rows -> attention in sA ----
    for (int i = wave; i < NF; i += 8) {
        float f1i = sf1[i];
        float x[4];
        #pragma unroll
        for (int t = 0; t < 4; ++t) {
            int j = lane + 32 * t;
            float v = f1i + sf2[j];
            v = v > 0.f ? v : 0.2f * v;                  // leaky relu
            v = (adj[i * NF + j] == 0) ? -1.0e9f : v;    // mask
            x[t] = v;
        }
        float mx = fmaxf(fmaxf(x[0], x[1]), fmaxf(x[2], x[3]));
        #pragma unroll
        for (int o = 16; o >= 1; o >>= 1) mx = fmaxf(mx, __shfl_xor(mx, o, 32));
        float s = 0.f;
        #pragma unroll
        for (int t = 0; t < 4; ++t) { x[t] = __expf(x[t] - mx); s += x[t]; }
        #pragma unroll
        for (int o = 16; o >= 1; o >>= 1) s += __shfl_xor(s, o, 32);
        float inv = 1.0f / s;
        #pragma unroll
        for (int t = 0; t < 4; ++t)
            sA[i * LDSTRIDE + lane + 32 * t] = x[t] * inv;
    }
    __syncthreads();

    // ---- GEMM2: h' = attention @ Wh, then ELU, store ----
    {
        const int mt = wave;
        const int m  = mt * 16 + l16;
        v8f acc[8];
        #pragma unroll
        for (int nt = 0; nt < 8; ++nt)
            #pragma unroll
            for (int r = 0; r < 8; ++r) acc[nt][r] = 0.f;
        for (int k0 = 0; k0 < NF; k0 += 4) {
            v2f a = *(const v2f*)(&sA[m * LDSTRIDE + k0 + kb]);        // attention row
            #pragma unroll
            for (int nt = 0; nt < 8; ++nt) {
                int n = nt * 16 + l16;
                // B fragment = Wh[k][n] = sWhT[n][k]: contiguous pair -> ds_load_b64
                v2f b = *(const v2f*)(&sWhT[n * LDSTRIDE + k0 + kb]);
                acc[nt] = __builtin_amdgcn_wmma_f32_16x16x4_f32(
                    false, a, false, b, (short)0, acc[nt], false, false);
            }
        }
        #pragma unroll
        for (int nt = 0; nt < 8; ++nt) {
            int n = nt * 16 + l16;
            #pragma unroll
            for (int r = 0; r < 8; ++r) {
                int row = mt * 16 + r + half * 8;
                float v = acc[nt][r];
                v = v > 0.f ? v : (__expf(v) - 1.0f);    // ELU (alpha=1)
                op[row * NF + n] = v;
            }
        }
    }
}

extern "C" void kernel_launch(void* const* d_in, const int* in_sizes, int n_in,
                              void* d_out, int out_size, void* d_ws, size_t ws_size,
                              hipStream_t stream) {
    const float* h   = (const float*)d_in[0];
    const float* Ww  = (const float*)d_in[1];
    const float* Wb  = (const float*)d_in[2];
    const float* a1  = (const float*)d_in[3];
    const float* a2  = (const float*)d_in[4];
    const float* ab  = (const float*)d_in[5];
    const int*   adj = (const int*)d_in[6];
    float* out = (float*)d_out;

    int n_bs = in_sizes[0] / (NF * NF);   // B*S = 1024
    gat_fused_kernel<<<dim3(n_bs), dim3(256), 0, stream>>>(h, Ww, Wb, a1, a2, ab, adj, out);
}